// WatermarkLayer_11819749998779
// MI455X (gfx1250) — compile-verified
//
#include <hip/hip_runtime.h>

#define LATENT      512
#define NSTEPS      50
#define ADAM_LR     0.01f
#define ADAM_B1     0.9f
#define ADAM_B2     0.999f
#define ADAM_EPS    1e-8f
// betaincinv(255.5, 0.5, 1-1e-5): upper tail 1-tau ~ 3e-13 (exact) or 2^-25
// (reference's float32-betainc bisection). sqrt(tau) rounds to 1.0f in f32
// either way, and the reference casts COS_THETA to mu.dtype (f32).
#define COS_THETA_F 1.0f

typedef __attribute__((ext_vector_type(2))) float f32x2;
typedef __attribute__((ext_vector_type(8))) float f32x8;

// Sum a per-lane float across all 32 lanes using V_WMMA_F32_16X16X4_F32.
// With B = all-ones, D[m][n] = sum_k A[m][k]; summing all 8 D VGPRs in a lane
// plus the xor-16 partner yields sum_m D[m][n] = sum of ALL 64 A entries,
// independent of the A-matrix lane layout. a.y = 0 so only the 32 partials
// contribute. Requires EXEC all-ones (callers keep branches wave-uniform).
__device__ __forceinline__ float wave_sum_wmma(float p) {
  f32x2 a; a[0] = p;    a[1] = 0.0f;
  f32x2 b; b[0] = 1.0f; b[1] = 1.0f;
  f32x8 cz = {0.f, 0.f, 0.f, 0.f, 0.f, 0.f, 0.f, 0.f};
  f32x8 d = __builtin_amdgcn_wmma_f32_16x16x4_f32(
      false, a, false, b, (short)0, cz, false, false);
  float s = ((d[0] + d[1]) + (d[2] + d[3])) + ((d[4] + d[5]) + (d[6] + d[7]));
  s += __shfl_xor(s, 16, 32);
  return s;
}

__global__ __launch_bounds__(256)
void wm_adam_kernel(const float* __restrict__ mu,
                    const float* __restrict__ logvar,
                    const float* __restrict__ carrier,
                    float* __restrict__ mu_out,
                    float* __restrict__ lv_out,
                    float* __restrict__ rowloss,
                    float* __restrict__ loss_accum,
                    int rows, float invB, int useWs) {
  const int lane = threadIdx.x & 31;
  const int wave = threadIdx.x >> 5;
  const int row  = blockIdx.x * 8 + wave;   // wave-uniform
  if (row >= rows) return;                  // whole wave exits together

  const size_t base = (size_t)row * LATENT;
  const float4* mu4  = (const float4*)(mu + base);
  const float4* lv4  = (const float4*)(logvar + base);
  float4*       lvo4 = (float4*)(lv_out + base);
  float4*       muo4 = (float4*)(mu_out + base);
  const float4* c4   = (const float4*)carrier;

  float x[16], cv[16], m[16], v[16];
#pragma unroll
  for (int j = 0; j < 4; ++j) {
    const int idx = lane + 32 * j;
    float4 tm = mu4[idx];
    float4 tc = c4[idx];
    lvo4[idx] = lv4[idx];                   // log_var passthrough
    x[4*j+0] = tm.x; x[4*j+1] = tm.y; x[4*j+2] = tm.z; x[4*j+3] = tm.w;
    cv[4*j+0] = tc.x; cv[4*j+1] = tc.y; cv[4*j+2] = tc.z; cv[4*j+3] = tc.w;
    m[4*j+0] = 0.f; m[4*j+1] = 0.f; m[4*j+2] = 0.f; m[4*j+3] = 0.f;
    v[4*j+0] = 0.f; v[4*j+1] = 0.f; v[4*j+2] = 0.f; v[4*j+3] = 0.f;
  }

  const float oneMB1 = 1.0f - ADAM_B1;
  const float oneMB2 = 1.0f - ADAM_B2;

  float lastLoss = 0.0f;
  for (int t = 1; t <= NSTEPS; ++t) {
    // dot / norm^2 partials; 2 accumulator chains each for ILP
    float pd0 = 0.f, pd1 = 0.f, pn0 = 0.f, pn1 = 0.f;
#pragma unroll
    for (int k = 0; k < 16; k += 2) {
      pd0 = fmaf(x[k],     cv[k],     pd0);
      pd1 = fmaf(x[k + 1], cv[k + 1], pd1);
      pn0 = fmaf(x[k],     x[k],      pn0);
      pn1 = fmaf(x[k + 1], x[k + 1],  pn1);
    }
    const float dot  = wave_sum_wmma(pd0 + pd1);
    const float nn   = wave_sum_wmma(pn0 + pn1);
    const float norm = sqrtf(nn);
    const float arg  = norm * COS_THETA_F - dot;
    if (t == NSTEPS) lastLoss = fmaxf(arg, 0.0f);   // loss recorded pre-update

    // grad of mean(relu(norm*cos - dot)) wrt mu, with 1/B folded in
    // (relu'(0) = 0, matching jax)
    const float actB  = (arg > 0.0f) ? invB : 0.0f;
    const float coefB = (arg > 0.0f) ? (COS_THETA_F * invB / norm) : 0.0f;

    // bias corrections are wave-uniform scalars (SALU-float); fold LR in
    const float corr1   = 1.0f / (1.0f - powf(ADAM_B1, (float)t));
    const float corr2   = 1.0f / (1.0f - powf(ADAM_B2, (float)t));
    const float lrCorr1 = ADAM_LR * corr1;

#pragma unroll
    for (int k = 0; k < 16; ++k) {
      const float g = fmaf(coefB, x[k], -actB * cv[k]);
      m[k] = fmaf(ADAM_B1, m[k], oneMB1 * g);
      v[k] = fmaf(ADAM_B2, v[k], oneMB2 * (g * g));
      const float num = m[k] * lrCorr1;               // LR * m_hat
      const float den = sqrtf(v[k] * corr2) + ADAM_EPS;
      // v_rcp_f32 (~1 ulp) instead of IEEE divide: ~6 fewer VALU ops/element
      x[k] -= num * __builtin_amdgcn_rcpf(den);
    }
  }

#pragma unroll
  for (int j = 0; j < 4; ++j) {
    float4 o;
    o.x = x[4*j+0]; o.y = x[4*j+1]; o.z = x[4*j+2]; o.w = x[4*j+3];
    muo4[lane + 32 * j] = o;
  }

  if (lane == 0) {
    if (useWs) rowloss[row] = lastLoss;
    else       atomicAdd(loss_accum, lastLoss * invB);
  }
}

__global__ __launch_bounds__(256)
void loss_reduce_kernel(const float* __restrict__ rowloss,
                        float* __restrict__ out, int n, float scale) {
  __shared__ float sh[256];
  const int tid = threadIdx.x;
  float s = 0.0f;
  for (int i = tid; i < n; i += 256) s += rowloss[i];   // fixed order
  sh[tid] = s;
  __syncthreads();
  if (tid < 128) sh[tid] += sh[tid + 128];
  __syncthreads();
  if (tid < 64) sh[tid] += sh[tid + 64];
  __syncthreads();
  if (tid < 32) {                      // wave 0 only: EXEC all-ones
    float p = sh[tid] + sh[tid + 32];
    float tot = wave_sum_wmma(p);
    if (tid == 0) out[0] = tot * scale;
  }
}

__global__ void zero_loss_kernel(float* p) {
  if (threadIdx.x == 0 && blockIdx.x == 0) p[0] = 0.0f;
}

extern "C" void kernel_launch(void* const* d_in, const int* in_sizes, int n_in,
                              void* d_out, int out_size, void* d_ws, size_t ws_size,
                              hipStream_t stream) {
  const float* mu      = (const float*)d_in[0];
  const float* logvar  = (const float*)d_in[1];
  const float* carrier = (const float*)d_in[2];

  const int rows = in_sizes[0] / LATENT;      // 131072
  const size_t N = (size_t)rows * LATENT;

  float* out      = (float*)d_out;
  float* mu_out   = out;                      // [rows, 512]
  float* lv_out   = out + N;                  // [rows, 512]
  float* loss_out = out + 2 * N;              // scalar

  const float invB = 1.0f / (float)rows;
  const int useWs = (ws_size >= (size_t)rows * sizeof(float)) ? 1 : 0;
  float* rowloss = (float*)d_ws;

  const int blocks = (rows + 7) / 8;          // 8 waves (rows) per 256-thread block

  if (!useWs) zero_loss_kernel<<<1, 64, 0, stream>>>(loss_out);
  wm_adam_kernel<<<blocks, 256, 0, stream>>>(mu, logvar, carrier, mu_out, lv_out,
                                             rowloss, loss_out, rows, invB, useWs);
  if (useWs)
    loss_reduce_kernel<<<1, 256, 0, stream>>>(rowloss, loss_out, rows, invB);
}